// UnsortedMambaLayer_63488206569950
// MI455X (gfx1250) — compile-verified
//
#include <hip/hip_runtime.h>
#include <math.h>

#define G     16
#define LSEQ  512
#define DM    256
#define DI    512
#define DS    16
#define RNK   16
#define NROWS (G * LSEQ)   /* 8192 dense rows */

typedef __attribute__((ext_vector_type(2))) float v2f;
typedef __attribute__((ext_vector_type(8))) float v8f;

__device__ __forceinline__ float siluf(float x)    { return x / (1.0f + __expf(-x)); }
__device__ __forceinline__ float sigmf(float x)    { return 1.0f / (1.0f + __expf(-x)); }
__device__ __forceinline__ float softplusf(float x){ return (x > 20.0f) ? x : log1pf(__expf(x)); }
__device__ __forceinline__ int   fliprow(int r, int f)
{ return f ? ((r / LSEQ) * LSEQ + (LSEQ - 1 - (r % LSEQ))) : r; }

// ---------------------------------------------------------------------------
// WMMA f32 GEMM:  C[m, j] (+)= sum_k A[m, k] * W[j, k]  (+ bias[j])
// W is row-major (N, K) with row stride ldw (covers all *_proj.T products).
// flipIn / flipOut reverse the time axis inside each graph (bwd direction).
// One wave computes a 64x16 strip: 4 M-subtiles share one B fragment per
// K-step (5 b64 loads : 4 WMMA), with fragments double-buffered in registers
// so the next step's loads are in flight while the current WMMAs execute.
// ---------------------------------------------------------------------------
__global__ __launch_bounds__(32)
void k_wmma_gemm(const float* __restrict__ A, int lda,
                 const float* __restrict__ W, int ldw,
                 const float* __restrict__ bias,
                 float* __restrict__ C, int ldc,
                 int K, int flipIn, int flipOut, int accum)
{
    const int n0   = blockIdx.x * 16;
    const int m0   = blockIdx.y * 64;
    const int lane = threadIdx.x;      // 0..31
    const int half = lane >> 4;        // 0 | 1
    const int l16  = lane & 15;

    const int nCol = n0 + l16;

    v8f acc[4];
    if (accum) {
        #pragma unroll
        for (int s = 0; s < 4; ++s)
            #pragma unroll
            for (int v = 0; v < 8; ++v) {
                int r = fliprow(m0 + s * 16 + v + 8 * half, flipOut);
                acc[s][v] = C[(long)r * ldc + nCol];
            }
    } else {
        #pragma unroll
        for (int s = 0; s < 4; ++s)
            #pragma unroll
            for (int v = 0; v < 8; ++v) acc[s][v] = 0.0f;
    }

    // Per-subtile A row pointers (per-graph sequence flip folded in here)
    const float* __restrict__ Arow[4];
    #pragma unroll
    for (int s = 0; s < 4; ++s) {
        int r = fliprow(m0 + s * 16 + l16, flipIn);
        Arow[s] = A + (long)r * lda;
    }
    const float* __restrict__ Wrow = W + (long)nCol * ldw;

    const int kkOff = 2 * half;   // lanes 0-15: K={k0,k0+1}; lanes 16-31: K={k0+2,k0+3}

    // ---- prologue: load fragments for k0 = 0 ----
    v2f bF, aF[4];
    bF.x = Wrow[kkOff]; bF.y = Wrow[kkOff + 1];
    #pragma unroll
    for (int s = 0; s < 4; ++s) { aF[s].x = Arow[s][kkOff]; aF[s].y = Arow[s][kkOff + 1]; }

    // ---- steady state: prefetch k0+4 while computing k0 ----
    for (int k0 = 0; k0 + 4 < K; k0 += 4) {
        const int kk = k0 + 4 + kkOff;
        v2f bN, aN[4];
        bN.x = Wrow[kk]; bN.y = Wrow[kk + 1];
        #pragma unroll
        for (int s = 0; s < 4; ++s) { aN[s].x = Arow[s][kk]; aN[s].y = Arow[s][kk + 1]; }

        #pragma unroll
        for (int s = 0; s < 4; ++s)
            acc[s] = __builtin_amdgcn_wmma_f32_16x16x4_f32(
                         false, aF[s], false, bF, (short)0, acc[s], false, false);

        bF = bN;
        #pragma unroll
        for (int s = 0; s < 4; ++s) aF[s] = aN[s];
    }

    // ---- epilogue: final K-step ----
    #pragma unroll
    for (int s = 0; s < 4; ++s)
        acc[s] = __builtin_amdgcn_wmma_f32_16x16x4_f32(
                     false, aF[s], false, bF, (short)0, acc[s], false, false);

    const float bv = bias ? bias[nCol] : 0.0f;
    #pragma unroll
    for (int s = 0; s < 4; ++s)
        #pragma unroll
        for (int v = 0; v < 8; ++v) {
            int r = fliprow(m0 + s * 16 + v + 8 * half, flipOut);
            C[(long)r * ldc + nCol] = acc[s][v] + bv;
        }
}

// ---------------------------------------------------------------------------
// Prologue: segment starts, dense scatter
// ---------------------------------------------------------------------------
__global__ void k_segstart(const int* __restrict__ batch, int* __restrict__ seg, int n)
{
    int i = blockIdx.x * blockDim.x + threadIdx.x;
    if (i >= n) return;
    if (i == 0 || batch[i] != batch[i - 1]) seg[batch[i]] = i;
}

__global__ void k_zero(float* __restrict__ p, int total)
{
    int i = blockIdx.x * blockDim.x + threadIdx.x;
    if (i < total) p[i] = 0.0f;
}

__global__ void k_scatter(const float* __restrict__ x, const int* __restrict__ batch,
                          const int* __restrict__ seg, float* __restrict__ dense, int nnodes)
{
    int i = blockIdx.x * blockDim.x + threadIdx.x;
    if (i >= nnodes * DM) return;
    int j = i % DM, n = i / DM;
    int b = batch[n], p = n - seg[b];
    dense[((long)b * LSEQ + p) * DM + j] = x[i];
}

// ---------------------------------------------------------------------------
// Depthwise causal conv (K=4) + bias + SiLU.  xz row stride 1024, xc = cols 0..511
// ---------------------------------------------------------------------------
__global__ void k_conv_silu(const float* __restrict__ xz, const float* __restrict__ cw,
                            const float* __restrict__ cb, float* __restrict__ xc, int total)
{
    int i = blockIdx.x * blockDim.x + threadIdx.x;
    if (i >= total) return;
    int c = i % DI, n = i / DI, t = n % LSEQ;
    float acc = cb[c];
    const float* w = cw + c * 4;     // conv_w (DI, 1, 4)
    #pragma unroll
    for (int j = 0; j < 4; ++j) {
        int tt = t - 3 + j;
        if (tt >= 0) acc += w[j] * xz[(long)(n - 3 + j) * 1024 + c];
    }
    xc[(long)n * DI + c] = siluf(acc);
}

// ---------------------------------------------------------------------------
// Selective scan: one lane per (graph, channel); 16 states live in VGPRs.
// dbc row = [dt_r(16) | B(16) | C(16)], dtl already includes dt_bias.
// ---------------------------------------------------------------------------
__global__ __launch_bounds__(256)
void k_scan(const float* __restrict__ xc, const float* __restrict__ dbc,
            const float* __restrict__ dtl, const float* __restrict__ A_log,
            const float* __restrict__ Dp, float* __restrict__ ys)
{
    int tid = blockIdx.x * blockDim.x + threadIdx.x;   // 0..8191
    int c = tid % DI, g = tid / DI;

    float A[DS];
    #pragma unroll
    for (int s = 0; s < DS; ++s) A[s] = -__expf(A_log[c * DS + s]);
    const float Dv = Dp[c];

    float h[DS];
    #pragma unroll
    for (int s = 0; s < DS; ++s) h[s] = 0.0f;

    for (int t = 0; t < LSEQ; ++t) {
        long n = (long)g * LSEQ + t;
        float u  = xc[n * DI + c];
        float dt = softplusf(dtl[n * DI + c]);
        const float* __restrict__ B  = dbc + n * 48 + RNK;
        const float* __restrict__ Cc = dbc + n * 48 + RNK + DS;
        float dtu = dt * u;
        float y = 0.0f;
        #pragma unroll
        for (int s = 0; s < DS; ++s) {
            float dA = __expf(dt * A[s]);
            h[s] = dA * h[s] + dtu * B[s];
            y += h[s] * Cc[s];
        }
        ys[n * DI + c] = y + u * Dv;
    }
}

// ys *= silu(z)  (z = xz columns 512..1023)
__global__ void k_gate_z(const float* __restrict__ xz, float* __restrict__ ys, int total)
{
    int i = blockIdx.x * blockDim.x + threadIdx.x;
    if (i >= total) return;
    int c = i % DI; long n = i / DI;
    ys[i] *= siluf(xz[n * 1024 + DI + c]);
}

// g = sigmoid(logits); yp = g*f + (1-g)*b   (in-place on logits buffer is fine)
__global__ void k_gate_combine(const float* __restrict__ gl, const float* __restrict__ f,
                               const float* __restrict__ b, float* __restrict__ yp, int total)
{
    int i = blockIdx.x * blockDim.x + threadIdx.x;
    if (i >= total) return;
    float g = sigmf(gl[i]);
    yp[i] = g * f[i] + (1.0f - g) * b[i];
}

__global__ void k_gather(const float* __restrict__ yp, const int* __restrict__ batch,
                         const int* __restrict__ seg, float* __restrict__ out, int nnodes)
{
    int i = blockIdx.x * blockDim.x + threadIdx.x;
    if (i >= nnodes * DM) return;
    int j = i % DM, n = i / DM;
    int b = batch[n], p = n - seg[b];
    out[i] = yp[((long)b * LSEQ + p) * DM + j];
}

// ---------------------------------------------------------------------------
struct Mparams {
    const float *in_proj, *conv_w, *conv_b, *x_proj, *dt_proj, *dt_bias, *A_log, *D, *out_proj;
};

extern "C" void kernel_launch(void* const* d_in, const int* in_sizes, int n_in,
                              void* d_out, int out_size, void* d_ws, size_t ws_size,
                              hipStream_t stream)
{
    (void)n_in; (void)out_size; (void)ws_size;

    const float* x; const int* batch;
    Mparams pf, pb; const float *gate_w, *gate_b;

    if (in_sizes[0] == NROWS * DM) {
        // setup_inputs() insertion order: x, batch, fwd{...}, bwd{...}, gate_w, gate_b
        x     = (const float*)d_in[0];
        batch = (const int*)  d_in[1];
        auto fill = [&](Mparams& p, int s) {
            p.in_proj  = (const float*)d_in[s + 0];
            p.conv_w   = (const float*)d_in[s + 1];
            p.conv_b   = (const float*)d_in[s + 2];
            p.x_proj   = (const float*)d_in[s + 3];
            p.dt_proj  = (const float*)d_in[s + 4];
            p.dt_bias  = (const float*)d_in[s + 5];
            p.A_log    = (const float*)d_in[s + 6];
            p.D        = (const float*)d_in[s + 7];
            p.out_proj = (const float*)d_in[s + 8];
        };
        fill(pf, 2); fill(pb, 11);
        gate_w = (const float*)d_in[20];
        gate_b = (const float*)d_in[21];
    } else {
        // jax tree_flatten sorted-key order: batch, bwd{sorted}, fwd{sorted}, gate_b, gate_w, x
        batch = (const int*)d_in[0];
        auto fill = [&](Mparams& p, int s) {
            p.A_log    = (const float*)d_in[s + 0];
            p.D        = (const float*)d_in[s + 1];
            p.conv_b   = (const float*)d_in[s + 2];
            p.conv_w   = (const float*)d_in[s + 3];
            p.dt_bias  = (const float*)d_in[s + 4];
            p.dt_proj  = (const float*)d_in[s + 5];
            p.in_proj  = (const float*)d_in[s + 6];
            p.out_proj = (const float*)d_in[s + 7];
            p.x_proj   = (const float*)d_in[s + 8];
        };
        fill(pb, 1); fill(pf, 10);
        gate_b = (const float*)d_in[19];
        gate_w = (const float*)d_in[20];
        x      = (const float*)d_in[21];
    }

    // ---- workspace carve (256B aligned slabs) ----
    char* wp = (char*)d_ws;
    auto carve = [&](size_t nfloats) {
        float* p = (float*)wp;
        wp += ((nfloats * sizeof(float) + 255) / 256) * 256;
        return p;
    };
    float* dense = carve((size_t)NROWS * DM);      // (8192, 256)
    float* xz    = carve((size_t)NROWS * 1024);    // (8192, 1024) per direction (reused)
    float* xc    = carve((size_t)NROWS * DI);      // conv+silu output
    float* dbc   = carve((size_t)NROWS * 48);      // [dt_r | B | C]
    float* dtl   = carve((size_t)NROWS * DI);      // dt linear (pre-softplus, bias added)
    float* ys    = carve((size_t)NROWS * DI);      // scan output -> gated in place
    float* fout  = carve((size_t)NROWS * DM);
    float* bout  = carve((size_t)NROWS * DM);
    float* gbuf  = carve((size_t)NROWS * DM);      // gate logits -> combined output
    int*   seg   = (int*)carve(64);

    const dim3 B256(256);
    const dim3 W32(32);
    const int elemDM = NROWS * DM;     // 2,097,152
    const int elemDI = NROWS * DI;     // 4,194,304
    const int MT64   = NROWS / 64;     // 128 M-strips per GEMM

    k_segstart<<<dim3((NROWS + 255) / 256), B256, 0, stream>>>(batch, seg, NROWS);
    k_zero    <<<dim3((elemDM + 255) / 256), B256, 0, stream>>>(dense, elemDM);
    k_scatter <<<dim3((elemDM + 255) / 256), B256, 0, stream>>>(x, batch, seg, dense, NROWS);

    for (int dir = 0; dir < 2; ++dir) {
        const Mparams& p = dir ? pb : pf;
        const int flip = dir;

        // xz = dense(flip) @ in_proj.T                (K=256, N=1024)
        k_wmma_gemm<<<dim3(1024 / 16, MT64), W32, 0, stream>>>(
            dense, DM, p.in_proj, DM, nullptr, xz, 1024, DM, flip, 0, 0);

        // causal depthwise conv + SiLU
        k_conv_silu<<<dim3((elemDI + 255) / 256), B256, 0, stream>>>(xz, p.conv_w, p.conv_b, xc, elemDI);

        // dbc = xc @ x_proj.T                          (K=512, N=48)
        k_wmma_gemm<<<dim3(48 / 16, MT64), W32, 0, stream>>>(
            xc, DI, p.x_proj, DI, nullptr, dbc, 48, DI, 0, 0, 0);

        // dtl = dt_r @ dt_proj.T + dt_bias             (K=16, N=512)
        k_wmma_gemm<<<dim3(DI / 16, MT64), W32, 0, stream>>>(
            dbc, 48, p.dt_proj, RNK, p.dt_bias, dtl, DI, RNK, 0, 0, 0);

        // selective scan (sequential over L, parallel over graph x channel)
        k_scan<<<dim3((G * DI) / 256), B256, 0, stream>>>(xc, dbc, dtl, p.A_log, p.D, ys);

        // ys *= silu(z)
        k_gate_z<<<dim3((elemDI + 255) / 256), B256, 0, stream>>>(xz, ys, elemDI);

        // out(dir) = ys @ out_proj.T, un-flip rows on store for bwd   (K=512, N=256)
        k_wmma_gemm<<<dim3(DM / 16, MT64), W32, 0, stream>>>(
            ys, DI, p.out_proj, DI, nullptr, dir ? bout : fout, DM, DI, 0, flip, 0);
    }

    // gate logits = [fout | bout] @ gate_w.T + gate_b  (split-K over the concat)
    k_wmma_gemm<<<dim3(DM / 16, MT64), W32, 0, stream>>>(
        fout, DM, gate_w, 2 * DM, nullptr, gbuf, DM, DM, 0, 0, 0);
    k_wmma_gemm<<<dim3(DM / 16, MT64), W32, 0, stream>>>(
        bout, DM, gate_w + DM, 2 * DM, gate_b, gbuf, DM, DM, 0, 0, 1);

    k_gate_combine<<<dim3((elemDM + 255) / 256), B256, 0, stream>>>(gbuf, fout, bout, gbuf, elemDM);
    k_gather      <<<dim3((elemDM + 255) / 256), B256, 0, stream>>>(gbuf, batch, seg, (float*)d_out, NROWS);
}